// ContrastiveLoss_44813688766886
// MI455X (gfx1250) — compile-verified
//
#include <hip/hip_runtime.h>

typedef __attribute__((ext_vector_type(2))) float v2f;
typedef __attribute__((ext_vector_type(8))) float v8f;

#define B 256
#define D 1024
#define C 64
#define LOG2E 1.4426950408889634f
#define LN2   0.6931471805599453f
#define EPSF  1e-18f

// ---------------------------------------------------------------------------
// Kernel 1: invk[j,d] = 1/sqrt(x^2 + eps) for both back_VF and back_AF.
// Pure streaming, 2 MB out; rsq via v_rsq_f32.
// ---------------------------------------------------------------------------
__global__ void invk_kernel(const float* __restrict__ V, const float* __restrict__ A,
                            float* __restrict__ invkV, float* __restrict__ invkA) {
    int idx = blockIdx.x * blockDim.x + threadIdx.x;   // 0 .. B*D-1
    float v = V[idx], a = A[idx];
    invkV[idx] = __builtin_amdgcn_rsqf(v * v + EPSF);
    invkA[idx] = __builtin_amdgcn_rsqf(a * a + EPSF);
}

// ---------------------------------------------------------------------------
// Kernel 2: row norms qnorm[i] = sqrt(sum_d x[i,d]^2 + eps), both tensors.
// One 256-thread block per row, LDS tree reduction.
// ---------------------------------------------------------------------------
__global__ void qnorm_kernel(const float* __restrict__ V, const float* __restrict__ A,
                             float* __restrict__ qnV, float* __restrict__ qnA) {
    const int i = blockIdx.x;
    const int t = threadIdx.x;                          // 0..255
    float sv = 0.f, sa = 0.f;
    for (int d = t; d < D; d += 256) {
        float v = V[i * D + d]; sv += v * v;
        float a = A[i * D + d]; sa += a * a;
    }
    __shared__ float lv[256], la[256];
    lv[t] = sv; la[t] = sa;
    __syncthreads();
    for (int s = 128; s > 0; s >>= 1) {
        if (t < s) { lv[t] += lv[t + s]; la[t] += la[t + s]; }
        __syncthreads();
    }
    if (t == 0) {
        qnV[i] = sqrtf(lv[0] + EPSF);
        qnA[i] = sqrtf(la[0] + EPSF);
    }
}

// ---------------------------------------------------------------------------
// Kernel 3: G = V @ A^T  (256x256, K=1024, f32) via V_WMMA_F32_16X16X4_F32.
// One wave per 16x16 tile.
//  A-operand (16x4): lane m=lane&15 is row M; half=lane>>4 selects K pair:
//    vgpr0 = V[ibase+m][k+2*half], vgpr1 = V[ibase+m][k+2*half+1]  -> float2 load
//  B-operand (4x16): lane m is col N; same float2 pattern from A rows.
//  C/D (16x16 f32): vgpr r = row ibase + r + 8*half, col jbase + m.
// ---------------------------------------------------------------------------
__global__ void gemm_wmma_kernel(const float* __restrict__ V, const float* __restrict__ A,
                                 float* __restrict__ G) {
    const int lane = threadIdx.x;        // 0..31 (wave32)
    const int half = lane >> 4;          // 0|1
    const int m    = lane & 15;
    const int ibase = blockIdx.y * 16;
    const int jbase = blockIdx.x * 16;

    const float* vrow = V + (size_t)(ibase + m) * D + 2 * half;
    const float* arow = A + (size_t)(jbase + m) * D + 2 * half;

    v8f acc = {};
    #pragma unroll 4
    for (int k = 0; k < D; k += 4) {
        v2f a = *(const v2f*)(vrow + k);
        v2f b = *(const v2f*)(arow + k);
        acc = __builtin_amdgcn_wmma_f32_16x16x4_f32(
            /*neg_a=*/false, a, /*neg_b=*/false, b,
            /*c_mod=*/(short)0, acc, /*reuse_a=*/false, /*reuse_b=*/false);
    }

    const int row0 = ibase + 8 * half;
    const int col  = jbase + m;
    #pragma unroll
    for (int r = 0; r < 8; ++r)
        G[(row0 + r) * B + col] = acc[r];
}

// ---------------------------------------------------------------------------
// Kernel 4: sim_pre over the C=64 axis; initializes d_out with 0.5*sim_pre.
// Coalesced: consecutive threads = consecutive d.
// ---------------------------------------------------------------------------
__global__ void sim_pre_kernel(const float* __restrict__ VF, const float* __restrict__ AF,
                               float* __restrict__ out) {
    int idx = blockIdx.x * blockDim.x + threadIdx.x;   // 0 .. B*D-1
    int b = idx >> 10;                                  // /D
    int d = idx & (D - 1);
    const float* v = VF + (size_t)b * C * D + d;
    const float* a = AF + (size_t)b * C * D + d;
    float num = 0.f, s1 = 0.f, s2 = 0.f;
    #pragma unroll 8
    for (int c = 0; c < C; ++c) {
        float x = v[c * D], y = a[c * D];
        num += x * y; s1 += x * x; s2 += y * y;
    }
    float denom = fmaxf(sqrtf(s1) * sqrtf(s2), EPSF);
    out[idx] = 0.5f * (num / denom);
}

// ---------------------------------------------------------------------------
// Kernel 5 (dominant): both log-terms fused.
//   negV[i,d] = sum_{j!=i} exp2(rV[i,j] * invkA[j,d])   (rV has log2e/qnormV folded)
//   negA[i,d] = sum_{j!=i} exp2(rA[i,j] * invkV[j,d])
//   out[i,d] -= 0.5 * ln2 * (log2(1+negV) + log2(1+negA))
// Block = 128 threads (one d each) x 8 i-values -> 16 v_exp_f32 per invk dword.
// rV/rA tiles staged in LDS (16 KB), broadcast reads in the j-loop.
// ---------------------------------------------------------------------------
#define DT 128
__global__ void logterm_kernel(const float* __restrict__ G,
                               const float* __restrict__ qnV, const float* __restrict__ qnA,
                               const float* __restrict__ invkV, const float* __restrict__ invkA,
                               float* __restrict__ out) {
    const int t     = threadIdx.x;             // 0..127
    const int d     = blockIdx.x * DT + t;     // 0..1023
    const int ibase = blockIdx.y * 8;          // 0..248

    __shared__ float rV[8][B];
    __shared__ float rA[8][B];
    for (int e = t; e < 8 * B; e += DT) {
        int ii = e >> 8, j = e & (B - 1);
        rV[ii][j] = G[(ibase + ii) * B + j] * (LOG2E / qnV[ibase + ii]);
        rA[ii][j] = G[j * B + (ibase + ii)] * (LOG2E / qnA[ibase + ii]);
    }
    __syncthreads();

    float accV[8] = {0.f, 0.f, 0.f, 0.f, 0.f, 0.f, 0.f, 0.f};
    float accA[8] = {0.f, 0.f, 0.f, 0.f, 0.f, 0.f, 0.f, 0.f};
    for (int j = 0; j < B; ++j) {
        float sA = invkA[j * D + d];
        float sV = invkV[j * D + d];
        #pragma unroll
        for (int ii = 0; ii < 8; ++ii) {
            accV[ii] += __builtin_amdgcn_exp2f(rV[ii][j] * sA);
            accA[ii] += __builtin_amdgcn_exp2f(rA[ii][j] * sV);
        }
    }

    #pragma unroll
    for (int ii = 0; ii < 8; ++ii) {
        const int i = ibase + ii;
        float selfA = invkA[i * D + d];
        float selfV = invkV[i * D + d];
        float negV = accV[ii] - __builtin_amdgcn_exp2f(rV[ii][i] * selfA);
        float negA = accA[ii] - __builtin_amdgcn_exp2f(rA[ii][i] * selfV);
        float term = __builtin_amdgcn_logf(1.0f + negV) +
                     __builtin_amdgcn_logf(1.0f + negA);
        out[i * D + d] -= 0.5f * LN2 * term;
    }
}

// ---------------------------------------------------------------------------
extern "C" void kernel_launch(void* const* d_in, const int* in_sizes, int n_in,
                              void* d_out, int out_size, void* d_ws, size_t ws_size,
                              hipStream_t stream) {
    const float* pre_VF  = (const float*)d_in[0];   // [256,64,1024]
    const float* pre_AF  = (const float*)d_in[1];   // [256,64,1024]
    const float* back_VF = (const float*)d_in[2];   // [256,1,1024] -> flat [256,1024]
    const float* back_AF = (const float*)d_in[3];   // [256,1,1024]
    float* out = (float*)d_out;                     // [256,1024]

    // Workspace layout (floats)
    float* ws    = (float*)d_ws;
    float* G     = ws;                      // 256*256          = 65536
    float* qnV   = ws + 65536;              // 256
    float* qnA   = ws + 65536 + 256;        // 256
    float* invkV = ws + 66048;              // 256*1024         = 262144
    float* invkA = ws + 66048 + 262144;     // 262144  (total ~2.26 MB)

    // 1) invk streams
    invk_kernel<<<(B * D) / 256, 256, 0, stream>>>(back_VF, back_AF, invkV, invkA);
    // 2) row norms
    qnorm_kernel<<<B, 256, 0, stream>>>(back_VF, back_AF, qnV, qnA);
    // 3) G = V @ A^T via f32 WMMA (16x16 grid of 1-wave tiles)
    gemm_wmma_kernel<<<dim3(B / 16, B / 16), 32, 0, stream>>>(back_VF, back_AF, G);
    // 4) 0.5*sim_pre -> out (initializes every output element)
    sim_pre_kernel<<<(B * D) / 256, 256, 0, stream>>>(pre_VF, pre_AF, out);
    // 5) fused log-terms, accumulate into out
    logterm_kernel<<<dim3(D / DT, B / 8), DT, 0, stream>>>(G, qnV, qnA, invkV, invkA, out);
}